// PersistenceLandscapeEncoder_16269336117487
// MI455X (gfx1250) — compile-verified
//
#include <hip/hip_runtime.h>
#include <stdint.h>

#define TPB 256          // threads per block for kernels 2/3 (8 wave32s)
#define TPB1 1024        // threads per block for the min/max reduction (one WGP)
#define NUM_L 5          // num_landscapes (reference setup)
#define RES 4096         // resolution     (reference setup)
#define LDS_PAIRS 2048   // pairs staged in LDS per tile (16 KB of float2)
#define MAX_CHUNKS 64    // default pair-dimension split (1024 blocks -> 8192 waves)

// Branchless sorted insert into a descending top-5 list (m0 >= m1 >= ... >= m4).
__device__ __forceinline__ void topk_insert(float v, float& m0, float& m1,
                                            float& m2, float& m3, float& m4) {
  float n0 = fmaxf(m0, v); v = fminf(m0, v); m0 = n0;
  float n1 = fmaxf(m1, v); v = fminf(m1, v); m1 = n1;
  float n2 = fmaxf(m2, v); v = fminf(m2, v); m2 = n2;
  float n3 = fmaxf(m3, v); v = fminf(m3, v); m3 = n3;
  m4 = fmaxf(m4, v);
}

// ---------------- Kernel 1: min(birth), max(death) ----------------
// One WGP-sized block; float4 loads pull two (birth,death) pairs per 16B.
__global__ void pl_minmax(const float4* __restrict__ pairs4, int n4, int nTailPairs,
                          const float2* __restrict__ pairsTail,
                          float* __restrict__ mm) {
  __shared__ float smin[TPB1];
  __shared__ float smax[TPB1];
  const int tid = threadIdx.x;
  float mn = 3.4e38f, mx = -3.4e38f;
  const float4* p = pairs4 + tid;
  for (int i = tid; i < n4; i += TPB1, p += TPB1) {
    float4 q = *p;                      // (b0, d0, b1, d1)
    mn = fminf(mn, fminf(q.x, q.z));
    mx = fmaxf(mx, fmaxf(q.y, q.w));
  }
  if (tid < nTailPairs) {               // odd trailing pair (none for N=32768)
    float2 q = pairsTail[tid];
    mn = fminf(mn, q.x);
    mx = fmaxf(mx, q.y);
  }
  smin[tid] = mn; smax[tid] = mx;
  __syncthreads();
  for (int s = TPB1 / 2; s > 0; s >>= 1) {
    if (tid < s) {
      smin[tid] = fminf(smin[tid], smin[tid + s]);
      smax[tid] = fmaxf(smax[tid], smax[tid + s]);
    }
    __syncthreads();
  }
  if (tid == 0) { mm[0] = smin[0]; mm[1] = smax[0]; }
}

// ---------------- Kernel 2: per-(t, pair-chunk) partial top-5 ----------------
// Grid: (RES/TPB, chunks). Each thread owns one t value; its chunk's pairs are
// staged into LDS via CDNA5 async loads (GLOBAL_LOAD_ASYNC_TO_LDS_B128, tracked
// by ASYNCcnt, drained with s_wait_asynccnt), then consumed as LDS broadcasts.
__global__ void pl_partial(const float2* __restrict__ pairs,
                           const float* __restrict__ mm,
                           float* __restrict__ partial,
                           int nPairs, int chunks) {
  __shared__ __align__(16) float2 sp[LDS_PAIRS];

  const int j     = blockIdx.x * TPB + threadIdx.x;  // t index
  const int chunk = blockIdx.y;
  const int per   = (nPairs + chunks - 1) / chunks;
  const int base  = chunk * per;
  const int count = min(per, nPairs - base);

  const float tmin = mm[0];
  const float tmax = mm[1];
  const float step = (tmax - tmin) / (float)(RES - 1);
  const float t    = tmin + step * (float)j;

  // Two interleaved top-5 accumulator sets halve the serial min/max dep chain.
  float a0 = 0.f, a1 = 0.f, a2 = 0.f, a3 = 0.f, a4 = 0.f;
  float b0 = 0.f, b1 = 0.f, b2 = 0.f, b3 = 0.f, b4 = 0.f;

  // Low 32 bits of a generic pointer to LDS == workgroup-relative LDS offset.
  const uint32_t ldsbase = (uint32_t)(uintptr_t)(&sp[0]);

  for (int tileBase = 0; tileBase < count; tileBase += LDS_PAIRS) {
    const int  tileCount  = min(LDS_PAIRS, count - tileBase);
    const int  totalBytes = tileCount * 8;
    const int  mainBytes  = totalBytes & ~15;
    const char* gsrc = (const char*)(pairs + base + tileBase);

    // Async copy global -> LDS, 16B per lane per issue (ASYNCcnt-tracked).
    for (int off = (int)threadIdx.x * 16; off < mainBytes; off += TPB * 16) {
      uint64_t ga = (uint64_t)(uintptr_t)(gsrc + off);
      uint32_t la = ldsbase + (uint32_t)off;
      asm volatile("global_load_async_to_lds_b128 %0, %1, off"
                   :: "v"(la), "v"(ga) : "memory");
    }
    if (threadIdx.x == 0 && (totalBytes & 15)) {  // 8-byte tail (one pair)
      uint64_t ga = (uint64_t)(uintptr_t)(gsrc + mainBytes);
      uint32_t la = ldsbase + (uint32_t)mainBytes;
      asm volatile("global_load_async_to_lds_b64 %0, %1, off"
                   :: "v"(la), "v"(ga) : "memory");
    }
    asm volatile("s_wait_asynccnt 0x0" ::: "memory");  // drain own async ops
    __syncthreads();                                   // publish LDS to all waves

    // Two pairs per ds_load_b128 broadcast (conflict-free: all lanes same addr).
    const float4* sp4 = (const float4*)sp;
    const int nQuad = tileCount >> 1;
    #pragma unroll 4
    for (int i = 0; i < nQuad; ++i) {
      float4 q = sp4[i];                // (b0, d0, b1, d1)
      float va = fmaxf(fminf(t - q.x, q.y - t), 0.0f);
      float vb = fmaxf(fminf(t - q.z, q.w - t), 0.0f);
      topk_insert(va, a0, a1, a2, a3, a4);
      topk_insert(vb, b0, b1, b2, b3, b4);
    }
    if (tileCount & 1) {
      float2 p = sp[tileCount - 1];
      float v = fmaxf(fminf(t - p.x, p.y - t), 0.0f);
      topk_insert(v, a0, a1, a2, a3, a4);
    }
    __syncthreads();  // everyone done reading before next tile's async writes
  }

  // Merge the two accumulator sets.
  topk_insert(b0, a0, a1, a2, a3, a4);
  topk_insert(b1, a0, a1, a2, a3, a4);
  topk_insert(b2, a0, a1, a2, a3, a4);
  topk_insert(b3, a0, a1, a2, a3, a4);
  topk_insert(b4, a0, a1, a2, a3, a4);

  if (j < RES) {
    float* dst = partial + ((size_t)chunk * RES + j) * NUM_L;
    dst[0] = a0; dst[1] = a1; dst[2] = a2; dst[3] = a3; dst[4] = a4;
  }
}

// ---------------- Kernel 3: merge chunk partials, write (5, RES) ----------------
__global__ void pl_merge(const float* __restrict__ partial,
                         float* __restrict__ out, int chunks) {
  const int j = blockIdx.x * blockDim.x + threadIdx.x;
  if (j >= RES) return;
  float m0 = 0.f, m1 = 0.f, m2 = 0.f, m3 = 0.f, m4 = 0.f;
  for (int c = 0; c < chunks; ++c) {
    const float* s = partial + ((size_t)c * RES + j) * NUM_L;
    topk_insert(s[0], m0, m1, m2, m3, m4);
    topk_insert(s[1], m0, m1, m2, m3, m4);
    topk_insert(s[2], m0, m1, m2, m3, m4);
    topk_insert(s[3], m0, m1, m2, m3, m4);
    topk_insert(s[4], m0, m1, m2, m3, m4);
  }
  out[0 * RES + j] = m0;
  out[1 * RES + j] = m1;
  out[2 * RES + j] = m2;
  out[3 * RES + j] = m3;
  out[4 * RES + j] = m4;
}

extern "C" void kernel_launch(void* const* d_in, const int* in_sizes, int n_in,
                              void* d_out, int out_size, void* d_ws, size_t ws_size,
                              hipStream_t stream) {
  (void)n_in; (void)out_size;
  const float2* pairs = (const float2*)d_in[0];
  const int nPairs = in_sizes[0] / 2;  // pairs stored as (N, 2) float32

  // Workspace layout: [0..1] = tmin/tmax floats; partials at +64 bytes.
  float* mm      = (float*)d_ws;
  float* partial = (float*)((char*)d_ws + 64);

  // Pick the largest pair-chunk count (parallelism) that fits the workspace.
  int chunks = MAX_CHUNKS;
  while (chunks > 1 &&
         (size_t)chunks * RES * NUM_L * sizeof(float) + 64 > ws_size)
    chunks >>= 1;
  if (chunks > nPairs) chunks = nPairs > 0 ? nPairs : 1;

  // Kernel 1: two pairs per float4; scalar tail for odd nPairs.
  const int n4 = nPairs / 2;
  const int nTail = nPairs - n4 * 2;
  pl_minmax<<<1, TPB1, 0, stream>>>((const float4*)pairs, n4, nTail,
                                    pairs + n4 * 2, mm);

  dim3 g2(RES / TPB, chunks);
  pl_partial<<<g2, TPB, 0, stream>>>(pairs, mm, partial, nPairs, chunks);

  pl_merge<<<RES / TPB, TPB, 0, stream>>>(partial, (float*)d_out, chunks);
}